// STFTProcessor_48576080118713
// MI455X (gfx1250) — compile-verified
//
#include <hip/hip_runtime.h>

typedef __attribute__((ext_vector_type(2))) float v2f;
typedef __attribute__((ext_vector_type(8))) float v8f;

#define N_FFT   4096
#define HOP     1024
#define PAD     2048
#define XLEN    524288
#define NFREQ   2049
#define NT      513
#define KC      32                 // K-chunk staged in LDS
#define NCHUNK  (N_FFT / KC)       // 128
#define SA      36                 // LDS row stride (floats) for basis tiles
#define ST2     66                 // LDS row stride (floats) per K-pair of frames tile
#define MT      128                // freq tile (8 waves x 16 rows)
#define NTT     32                 // time tile (each wave covers both 16-col halves)

__global__ __launch_bounds__(256, 2)
void stft_wmma_f32(const float* __restrict__ xg,
                   const float* __restrict__ win,
                   const float* __restrict__ cb,
                   const float* __restrict__ sb,
                   float* __restrict__ out)
{
    __shared__ __align__(16) float sAc[MT * SA];        // windowed cos tile, [m][k]
    __shared__ __align__(16) float sAs[MT * SA];        // windowed sin tile, [m][k]
    __shared__ __align__(16) float sF [(KC / 2) * ST2]; // frames, [k/2][t][k&1] packed pairs

    const int tid  = threadIdx.x;
    const int lane = tid & 31;
    const int wave = tid >> 5;

    const int bc = blockIdx.z;                 // 0..15  (b*2 + c)
    const int t0 = blockIdx.x * NTT;           // frame-tile origin
    const int f0 = blockIdx.y * MT;            // freq-tile origin

    const float* xb = xg + (size_t)bc * XLEN;

    // ---- staging mapping: tid -> (k within chunk, row group) ----
    const int sk  = tid & 31;                  // K index in chunk (coalesced axis)
    const int srw = tid >> 5;                  // row group 0..7
    const int sfo = (sk >> 1) * ST2 + (sk & 1);// frames LDS base for this thread's k

    // ---- WMMA compute mapping ----
    const int msub = wave << 4;                // wave's 16-row freq stripe (0..112)
    const int am   = lane & 15;                // A: M = am
    const int koff = (lane >> 4) << 1;         // A: K offset 0 or 2
    const int rowA = (msub + am) * SA;         // even => 8B-aligned v2f loads
    const int tc0  = lane & 15;                // B col half 0 (half 1 = +16)
    const int rowB = (lane >> 4) * ST2 + 2 * tc0; // B: packed K-pair base (even => 8B aligned)

    v8f accC0 = {}, accS0 = {};                // cos/sin, time cols 0..15
    v8f accC1 = {}, accS1 = {};                // cos/sin, time cols 16..31

    float rF[4], rC[16], rS[16];

    // ---------------- prologue: stage chunk 0 ----------------
    {
        const int n = sk;
        const float w = win[n];
#pragma unroll
        for (int i = 0; i < 4; ++i) {
            int t = t0 + srw + 8 * i;
            int p = t * HOP + n - PAD;
            p = (p < 0) ? -p : p;
            p = (p >= XLEN) ? (2 * XLEN - 2 - p) : p;
            rF[i] = xb[p];
        }
#pragma unroll
        for (int i = 0; i < 16; ++i) {
            int f = f0 + srw + 8 * i;
            float c = 0.0f, s = 0.0f;
            if (f < NFREQ) {
                size_t o = (size_t)f * N_FFT + n;
                c = cb[o] * w;
                s = sb[o] * w;
            }
            rC[i] = c; rS[i] = s;
        }
#pragma unroll
        for (int i = 0; i < 4; ++i) sF[sfo + 2 * (srw + 8 * i)] = rF[i];
#pragma unroll
        for (int i = 0; i < 16; ++i) {
            sAc[(srw + 8 * i) * SA + sk] = rC[i];
            sAs[(srw + 8 * i) * SA + sk] = rS[i];
        }
    }
    __syncthreads();

    // ---------------- main K loop: 128 chunks of 32 ----------------
    for (int ch = 0; ch < NCHUNK; ++ch) {
        // prefetch next chunk into registers while WMMAs run on current LDS tile
        if (ch + 1 < NCHUNK) {
            const int n = (ch + 1) * KC + sk;
            const float w = win[n];
#pragma unroll
            for (int i = 0; i < 4; ++i) {
                int t = t0 + srw + 8 * i;
                int p = t * HOP + n - PAD;
                p = (p < 0) ? -p : p;
                p = (p >= XLEN) ? (2 * XLEN - 2 - p) : p;
                rF[i] = xb[p];
            }
#pragma unroll
            for (int i = 0; i < 16; ++i) {
                int f = f0 + srw + 8 * i;
                float c = 0.0f, s = 0.0f;
                if (f < NFREQ) {
                    size_t o = (size_t)f * N_FFT + n;
                    c = cb[o] * w;
                    s = sb[o] * w;
                }
                rC[i] = c; rS[i] = s;
            }
        }

        // 8 K-steps x 4 WMMAs: A(cos)/A(sin) reused across both time halves,
        // B operands are single 8B packed-pair LDS loads (no register packing)
#pragma unroll
        for (int kk = 0; kk < KC; kk += 4) {
            v2f ac = *(const v2f*)(sAc + rowA + kk + koff);
            v2f as = *(const v2f*)(sAs + rowA + kk + koff);
            const int rb = rowB + (kk >> 1) * ST2;
            v2f b0 = *(const v2f*)(sF + rb);
            v2f b1 = *(const v2f*)(sF + rb + 32);
            accC0 = __builtin_amdgcn_wmma_f32_16x16x4_f32(
                        false, ac, false, b0, (short)0, accC0, false, false);
            accS0 = __builtin_amdgcn_wmma_f32_16x16x4_f32(
                        false, as, false, b0, (short)0, accS0, false, false);
            accC1 = __builtin_amdgcn_wmma_f32_16x16x4_f32(
                        false, ac, false, b1, (short)0, accC1, false, false);
            accS1 = __builtin_amdgcn_wmma_f32_16x16x4_f32(
                        false, as, false, b1, (short)0, accS1, false, false);
        }

        __syncthreads();   // all waves done reading current tile
        if (ch + 1 < NCHUNK) {
#pragma unroll
            for (int i = 0; i < 4; ++i) sF[sfo + 2 * (srw + 8 * i)] = rF[i];
#pragma unroll
            for (int i = 0; i < 16; ++i) {
                sAc[(srw + 8 * i) * SA + sk] = rC[i];
                sAs[(srw + 8 * i) * SA + sk] = rS[i];
            }
        }
        __syncthreads();   // next tile visible before compute
    }

    // ---------------- epilogue: guarded stores ----------------
    const size_t plane = (size_t)NFREQ * NT;                      // 2049*513
    float* outR = out + (size_t)bc * plane;                       // real block
    float* outI = out + (size_t)16 * plane + (size_t)bc * plane;  // imag block
    const int tg0 = t0 + tc0;
    const int tg1 = tg0 + 16;
    const int fb  = f0 + msub + ((lane >> 4) << 3);               // +8 for lanes 16..31
#pragma unroll
    for (int v = 0; v < 8; ++v) {
        int f = fb + v;
        if (f < NFREQ) {
            size_t rowo = (size_t)f * NT;
            if (tg0 < NT) {
                outR[rowo + tg0] = accC0[v];
                outI[rowo + tg0] = accS0[v];
            }
            if (tg1 < NT) {
                outR[rowo + tg1] = accC1[v];
                outI[rowo + tg1] = accS1[v];
            }
        }
    }
}

extern "C" void kernel_launch(void* const* d_in, const int* in_sizes, int n_in,
                              void* d_out, int out_size, void* d_ws, size_t ws_size,
                              hipStream_t stream) {
    // setup_inputs order: x, window, cos_basis, sin_basis  (all fp32)
    const float* x   = (const float*)d_in[0];
    const float* win = (const float*)d_in[1];
    const float* cb  = (const float*)d_in[2];
    const float* sb  = (const float*)d_in[3];
    float* out = (float*)d_out;

    dim3 grid((NT + NTT - 1) / NTT,      // 17 time tiles
              (NFREQ + MT - 1) / MT,     // 17 freq tiles
              16);                       // b*c slices
    stft_wmma_f32<<<grid, dim3(256), 0, stream>>>(x, win, cb, sb, out);

    (void)in_sizes; (void)n_in; (void)out_size; (void)d_ws; (void)ws_size;
}